// QuantumKernelRegressionGraph_65481071398496
// MI455X (gfx1250) — compile-verified
//
#include <hip/hip_runtime.h>

// CDNA5 / gfx1250, wave32.
typedef float v2f __attribute__((ext_vector_type(2)));
typedef float v8f __attribute__((ext_vector_type(8)));

// out[i] = sum_j relu(diag_i * W1[j] + b1[j]) * W2[j] + b2, with
// diag_i = exp(-gamma * (xn - 2*xn + xn)) == 1.0f EXACTLY in fp32
// (xn and the matmul diagonal are the same fp32 product; v-2v+v == +0;
//  exp(+-0) == 1 for the finite gamma provided). So the output is one
// scalar broadcast over N rows. X and gamma are dead inputs.
//
// The 32-wide head reduction is done on the matrix pipe with the exact-fp32
// V_WMMA_F32_16X16X4_F32: A[m,k] = p[4m+k] (m=0..7, rows 8..15 zero),
// B = all-ones (constant matrix => layout-independent), so
// D[m,n] = sum_k p[4m+k]; summing D VGPRs 0..7 yields S in lanes 0..15.
__global__ __launch_bounds__(256) void qkr_broadcast_kernel(
    const float* __restrict__ W1,   // [1,32]
    const float* __restrict__ b1,   // [32]
    const float* __restrict__ W2,   // [32,1]
    const float* __restrict__ b2,   // [1]
    float* __restrict__ out, int n)
{
  const int lane = threadIdx.x & 31;

  // p[j] = relu(1.0*W1[j] + b1[j]) * W2[j], one element per lane (H == 32).
  const float diag = 1.0f;
  float p = fmaxf(fmaf(diag, W1[lane], b1[lane]), 0.0f) * W2[lane];

  // Pack p into the fp32 16x4 A-matrix layout:
  //   lane m      : VGPR0 = A[m,0] = p[4m+0], VGPR1 = A[m,1] = p[4m+1]
  //   lane m + 16 : VGPR0 = A[m,2] = p[4m+2], VGPR1 = A[m,3] = p[4m+3]
  // Rows m = 8..15 are zero. Shuffles run with all lanes active (EXEC all-1s),
  // masking happens afterwards via cndmask, so ds_bpermute sources are valid.
  const int  m  = lane & 15;
  const bool hi = lane >= 16;
  const int  s0 = (4 * m + (hi ? 2 : 0)) & 31;
  const int  s1 = (4 * m + (hi ? 3 : 1)) & 31;
  float t0 = __shfl(p, s0, 32);
  float t1 = __shfl(p, s1, 32);

  v2f A;
  A.x = (m < 8) ? t0 : 0.0f;
  A.y = (m < 8) ? t1 : 0.0f;
  v2f B;
  B.x = 1.0f;
  B.y = 1.0f;                      // all-ones 4x16 B
  v8f C = {};

  // D = A x B + 0  (exact fp32 matrix pipe)
  C = __builtin_amdgcn_wmma_f32_16x16x4_f32(
      /*neg_a=*/false, A, /*neg_b=*/false, B,
      /*c_mod=*/(short)0, C, /*reuse_a=*/false, /*reuse_b=*/false);

  // Lanes 0..15 hold D[m=r, n=lane] in VGPR r => sum the 8 row-partials.
  float S = ((C[0] + C[1]) + (C[2] + C[3])) + ((C[4] + C[5]) + (C[6] + C[7]));
  S = __shfl(S, 0, 32);            // broadcast lane 0's full sum
  const float outv = S + b2[0];

  // Coalesced float4 broadcast store: 16 blocks x 256 threads covers 16384.
  const int i = (int)(blockIdx.x * blockDim.x + threadIdx.x) * 4;
  if (i + 3 < n) {
    float4 v = make_float4(outv, outv, outv, outv);
    *reinterpret_cast<float4*>(out + i) = v;
  } else {
    for (int k = i; k < n; ++k) out[k] = outv;   // tail (unused for n=16384)
  }
}

extern "C" void kernel_launch(void* const* d_in, const int* in_sizes, int n_in,
                              void* d_out, int out_size, void* d_ws, size_t ws_size,
                              hipStream_t stream) {
  // setup_inputs order: 0=X (dead), 1=gamma (dead), 2=W1, 3=b1, 4=W2, 5=b2
  const float* W1 = (const float*)d_in[2];
  const float* b1 = (const float*)d_in[3];
  const float* W2 = (const float*)d_in[4];
  const float* b2 = (const float*)d_in[5];
  float* out = (float*)d_out;

  const int n       = out_size;            // 16384
  const int threads = 256;                 // 8 wave32s / block
  const int vecs    = (n + 3) / 4;         // float4 slots
  const int blocks  = (vecs + threads - 1) / threads;  // 16 blocks

  qkr_broadcast_kernel<<<blocks, threads, 0, stream>>>(W1, b1, W2, b2, out, n);
}